// GraphSAGEWithAttention_89876485636274
// MI455X (gfx1250) — compile-verified
//
#include <hip/hip_runtime.h>
#include <hip/hip_bf16.h>
#include <math.h>

typedef __attribute__((ext_vector_type(2))) float v2f;
typedef __attribute__((ext_vector_type(8))) float v8f;

#define N_NODES 50000
#define N_EDGES 800000
#define E_TOT   (N_EDGES + N_NODES)
#define NEG_SLOPE 0.2f

// ---------------------------------------------------------------------------
// GEMM C[N,O] = A[N,K] * B[K,O], fp32 WMMA (V_WMMA_F32_16X16X4_F32).
// One wave per 16x16 output tile. Requires N%16==0, K%4==0, O%16==0 (true here).
// A frag (16x4): lanes 0-15 -> M=lane, K=kk+{0,1}; lanes 16-31 -> K=kk+{2,3}.
// B frag (4x16): lanes 0-15 -> K=kk+{0,1}, N=lane; lanes 16-31 -> K=kk+{2,3}.
// C/D (16x16):  vgpr r, lanes 0-15 -> M=r; lanes 16-31 -> M=r+8.
// ---------------------------------------------------------------------------
__global__ void gemm_wmma_f32(const float* __restrict__ A, const float* __restrict__ B,
                              float* __restrict__ C, int n, int k, int o) {
  const int lane  = threadIdx.x & 31;
  const int wave  = threadIdx.x >> 5;
  const int tile  = blockIdx.x * (blockDim.x >> 5) + wave;
  const int ctiles = o >> 4;
  const int ntiles = (n >> 4) * ctiles;
  if (tile >= ntiles) return;                 // wave-uniform -> EXEC stays all-1
  const int tm = tile / ctiles;
  const int tn = tile % ctiles;

  const int mrow = lane & 15;
  const int koff = (lane >> 4) << 1;          // 0 for lanes 0-15, 2 for 16-31
  const float* arow = A + (size_t)(tm * 16 + mrow) * k;
  const float* bcol = B + tn * 16 + (lane & 15);

  v8f c = {};
  for (int kk = 0; kk < k; kk += 4) {
    v2f a, b;
    a.x = arow[kk + koff + 0];
    a.y = arow[kk + koff + 1];
    b.x = bcol[(size_t)(kk + koff + 0) * o];
    b.y = bcol[(size_t)(kk + koff + 1) * o];
    c = __builtin_amdgcn_wmma_f32_16x16x4_f32(false, a, false, b, (short)0, c,
                                              false, false);
  }

  const int rbase = tm * 16 + ((lane >> 4) << 3);   // +8 rows for upper lanes
  float* crow = C + tn * 16 + (lane & 15);
#pragma unroll
  for (int r = 0; r < 8; ++r)
    crow[(size_t)(rbase + r) * o] = c[r];
}

// ---------------------------------------------------------------------------
// Per (node, head): e_s = <h, a_s>, e_d = <h, a_d>; also init m=-inf, z=0.
// ---------------------------------------------------------------------------
__global__ void attn_coef(const float* __restrict__ h, const float* __restrict__ as,
                          const float* __restrict__ ad, float* __restrict__ es,
                          float* __restrict__ ed, float* __restrict__ m,
                          float* __restrict__ z, int n, int heads, int c) {
  int i = blockIdx.x * blockDim.x + threadIdx.x;
  if (i >= n * heads) return;
  int node = i / heads, hd = i % heads;
  const float* hp  = h  + (size_t)node * heads * c + (size_t)hd * c;
  const float* asp = as + hd * c;
  const float* adp = ad + hd * c;
  float s0 = 0.f, s1 = 0.f;
  for (int j = 0; j < c; ++j) { float v = hp[j]; s0 += v * asp[j]; s1 += v * adp[j]; }
  es[i] = s0;
  ed[i] = s1;
  m[i]  = -1e30f;
  z[i]  = 0.f;
}

// float atomic max via monotonic int mapping (native int atomics, no CAS loop)
__device__ __forceinline__ void atomicMaxF(float* addr, float v) {
  if (v >= 0.f) atomicMax((int*)addr, __float_as_int(v));
  else          atomicMin((unsigned int*)addr, __float_as_uint(v));
}

__device__ __forceinline__ void edge_sd(const long long* __restrict__ ei, int e,
                                        int& s, int& d) {
  if (e < N_EDGES) { s = (int)ei[e]; d = (int)ei[N_EDGES + e]; }
  else             { s = e - N_EDGES; d = s; }  // self loops appended
}

// pass 1: leaky-relu logits per (edge, head), segment max into m[dst]
__global__ void edge_max(const long long* __restrict__ ei, const float* __restrict__ es,
                         const float* __restrict__ ed, float* __restrict__ m,
                         float* __restrict__ ee, int heads) {
  int e = blockIdx.x * blockDim.x + threadIdx.x;
  if (e >= E_TOT) return;
  int s, d; edge_sd(ei, e, s, d);
  for (int h = 0; h < heads; ++h) {
    float v = es[s * heads + h] + ed[d * heads + h];
    v = (v >= 0.f) ? v : NEG_SLOPE * v;
    ee[(size_t)e * heads + h] = v;
    atomicMaxF(&m[d * heads + h], v);
  }
}

// pass 2: p = exp(e - m[dst]); segment sum into z[dst]; ee := p
__global__ void edge_expsum(const long long* __restrict__ ei, const float* __restrict__ m,
                            float* __restrict__ z, float* __restrict__ ee, int heads) {
  int e = blockIdx.x * blockDim.x + threadIdx.x;
  if (e >= E_TOT) return;
  int s, d; edge_sd(ei, e, s, d);
  for (int h = 0; h < heads; ++h) {
    float p = __expf(ee[(size_t)e * heads + h] - m[d * heads + h]);
    ee[(size_t)e * heads + h] = p;
    atomicAdd(&z[d * heads + h], p);
  }
}

// pass 3: out[dst] += alpha * h[src]; block = one edge, thread = one channel
__global__ void edge_scatter(const long long* __restrict__ ei, const float* __restrict__ hbuf,
                             const float* __restrict__ ee, const float* __restrict__ z,
                             float* __restrict__ out, int heads, int c) {
  int e = blockIdx.x;
  int j = threadIdx.x;           // 0 .. heads*c-1
  int s, d; edge_sd(ei, e, s, d);
  int h = j / c;
  int hc = heads * c;
  float alpha = ee[(size_t)e * heads + h] / (z[d * heads + h] + 1e-16f);
  atomicAdd(&out[(size_t)d * hc + j], alpha * hbuf[(size_t)s * hc + j]);
}

// out += bias (broadcast over channels), optional ELU
__global__ void bias_act(float* __restrict__ buf, const float* __restrict__ b,
                         long long total, int o, int do_elu) {
  long long i = (long long)blockIdx.x * blockDim.x + threadIdx.x;
  if (i >= total) return;
  float v = buf[i] + b[(int)(i % o)];
  if (do_elu) v = (v > 0.f) ? v : (__expf(v) - 1.f);
  buf[i] = v;
}

// final linear classifier: out[n] = <h[n,:], Wc> + bc
__global__ void classifier(const float* __restrict__ h, const float* __restrict__ wc,
                           const float* __restrict__ bc, float* __restrict__ out,
                           int n, int c) {
  int i = blockIdx.x * blockDim.x + threadIdx.x;
  if (i >= n) return;
  const float* hp = h + (size_t)i * c;
  float s = 0.f;
  for (int j = 0; j < c; ++j) s += hp[j] * wc[j];
  out[i] = s + bc[0];
}

// ---------------------------------------------------------------------------
extern "C" void kernel_launch(void* const* d_in, const int* in_sizes, int n_in,
                              void* d_out, int out_size, void* d_ws, size_t ws_size,
                              hipStream_t stream) {
  (void)in_sizes; (void)n_in; (void)out_size; (void)ws_size;
  const float*     x   = (const float*)d_in[0];
  const long long* ei  = (const long long*)d_in[1];   // int64 [2, 800000]
  const float* W1  = (const float*)d_in[2];
  const float* a1s = (const float*)d_in[3];
  const float* a1d = (const float*)d_in[4];
  const float* b1  = (const float*)d_in[5];
  const float* W2  = (const float*)d_in[6];
  const float* a2s = (const float*)d_in[7];
  const float* a2d = (const float*)d_in[8];
  const float* b2  = (const float*)d_in[9];
  const float* W3  = (const float*)d_in[10];
  const float* a3s = (const float*)d_in[11];
  const float* a3d = (const float*)d_in[12];
  const float* b3  = (const float*)d_in[13];
  const float* Wc  = (const float*)d_in[14];
  const float* bc  = (const float*)d_in[15];

  // workspace layout (floats): ~119 MB
  float* ws   = (float*)d_ws;
  float* hbuf = ws;                                   // 50000*256
  float* obuf = hbuf + (size_t)N_NODES * 256;         // 50000*256
  float* es   = obuf + (size_t)N_NODES * 256;         // 50000*4
  float* ed   = es + (size_t)N_NODES * 4;
  float* m    = ed + (size_t)N_NODES * 4;
  float* z    = m  + (size_t)N_NODES * 4;
  float* ee   = z  + (size_t)N_NODES * 4;             // 850000*4

  auto layer = [&](const float* in, int K, int O, int heads, const float* W,
                   const float* as, const float* ad, const float* b, int do_elu) {
    const int C = O / heads;
    const int tiles = (N_NODES / 16) * (O / 16);
    gemm_wmma_f32<<<(tiles + 3) / 4, 128, 0, stream>>>(in, W, hbuf, N_NODES, K, O);
    const int nh = N_NODES * heads;
    attn_coef<<<(nh + 255) / 256, 256, 0, stream>>>(hbuf, as, ad, es, ed, m, z,
                                                    N_NODES, heads, C);
    hipMemsetAsync(obuf, 0, (size_t)N_NODES * O * sizeof(float), stream);
    edge_max<<<(E_TOT + 255) / 256, 256, 0, stream>>>(ei, es, ed, m, ee, heads);
    edge_expsum<<<(E_TOT + 255) / 256, 256, 0, stream>>>(ei, m, z, ee, heads);
    edge_scatter<<<E_TOT, O, 0, stream>>>(ei, hbuf, ee, z, obuf, heads, C);
    bias_act<<<(int)(((long long)N_NODES * O + 255) / 256), 256, 0, stream>>>(
        obuf, b, (long long)N_NODES * O, O, do_elu);
  };

  layer(x,    512, 256, 4, W1, a1s, a1d, b1, 1);   // conv1 + ELU
  layer(obuf, 256, 128, 4, W2, a2s, a2d, b2, 1);   // conv2 + ELU
  layer(obuf, 128,  64, 1, W3, a3s, a3d, b3, 0);   // conv3 (1 head, mean==identity)

  classifier<<<(N_NODES + 255) / 256, 256, 0, stream>>>(obuf, Wc, bc, (float*)d_out,
                                                        N_NODES, 64);
}